// VoronoiPropagation_34969623724658
// MI455X (gfx1250) — compile-verified
//
#include <hip/hip_runtime.h>
#include <stdint.h>

#define BATCH   16
#define Hh      512
#define Ww      512
#define NCLUST  64
#define NEIGHR  10
#define WINS    20
#define NITER   50
#define GRADW   10.0f
#define COLORW  10.0f

#define HW   (Hh * Ww)         /* 262144 = 2^18 */
#define BHW  (BATCH * HW)      /* 4194304 */

typedef __attribute__((ext_vector_type(2))) float v2f;
typedef __attribute__((ext_vector_type(8))) float v8f;

#if __has_builtin(__builtin_amdgcn_wmma_f32_16x16x4_f32)
#define HAVE_WMMA_F32X4 1
#else
#define HAVE_WMMA_F32X4 0
#endif

// ---------------------------------------------------------------------------
// CDNA5 async gather into LDS: per-lane global addr -> per-lane LDS slot.
// ---------------------------------------------------------------------------
typedef __attribute__((address_space(1))) int GINT;   // global
typedef __attribute__((address_space(3))) int LINT;   // LDS

__device__ __forceinline__ void async_gather_b32(void* lds_dst, const void* gsrc) {
#if __has_builtin(__builtin_amdgcn_global_load_async_to_lds_b32)
  __builtin_amdgcn_global_load_async_to_lds_b32((GINT*)gsrc, (LINT*)lds_dst, 0, 0);
#else
  unsigned lds_off = (unsigned)(size_t)lds_dst;
  asm volatile("global_load_async_to_lds_b32 %0, %1, off"
               :: "v"(lds_off), "v"(gsrc)
               : "memory");
#endif
}

__device__ __forceinline__ void wait_async0() {
#if __has_builtin(__builtin_amdgcn_s_wait_asynccnt)
  __builtin_amdgcn_s_wait_asynccnt(0);
#else
  asm volatile("s_wait_asynccnt 0" ::: "memory");
#endif
}

// Band matrices for the separable Sobel as matmuls.
// Av[m][r]: vertical smooth [1,2,1];  Bv[m][r]: vertical diff [-1,0,1].
__device__ __forceinline__ float band_smooth(int m, int r) {
  if (r == m || r == m + 2) return 1.0f;
  if (r == m + 1) return 2.0f;
  return 0.0f;
}
__device__ __forceinline__ float band_diff(int m, int r) {
  if (r == m) return -1.0f;
  if (r == m + 2) return 1.0f;
  return 0.0f;
}

// ---------------------------------------------------------------------------
// Kernel 1: fused grayscale + Sobel via WMMA -> grad_map and wgrad = grad^4*10
// One wave per 16x16 output tile; 18x18 haloed gray tile in LDS.
//   gx = (Av x G)[:,2:18] - (Av x G)[:,0:16]
//   gy = (Bv x G)[:,0:16] + 2*(Bv x G)[:,1:17] + (Bv x G)[:,2:18]
// K = 18 accumulated as 5 chunks of K=4 (zero padded), all windows share the
// same C/D register layout -> elementwise combine, no shuffles.
// ---------------------------------------------------------------------------
__global__ __launch_bounds__(32) void k_preprocess(const float* __restrict__ x,
                                                   float* __restrict__ grad_out,
                                                   float* __restrict__ wgrad) {
  __shared__ float g[20][19];   // rows 18,19 zero padding for the K=20 sweep
  const int b  = blockIdx.z;
  const int ty = blockIdx.y * 16, tx = blockIdx.x * 16;
  const int lane = threadIdx.x;          // single wave32

  const float* xr = x + (size_t)b * 3 * HW;
  const float* xg = xr + HW;
  const float* xb = xg + HW;

  // prefetch tile rows (emits global_prefetch_b8)
  __builtin_prefetch(xr + (size_t)ty * Ww + tx, 0, 1);

  for (int i = lane; i < 20 * 18; i += 32) {   // [r<20][c<18]
    int r = i / 18, c = i - r * 18;
    int gy = ty + r - 1, gx = tx + c - 1;
    float v = 0.0f;
    if (r < 18 && gy >= 0 && gy < Hh && gx >= 0 && gx < Ww) {
      int p = gy * Ww + gx;
      v = 0.2989f * xr[p] + 0.587f * xg[p] + 0.114f * xb[p];
    }
    g[r][c] = v;
  }
  __syncthreads();

  const int nn   = lane & 15;   // N (column) for B/C/D operands
  const int half = lane >> 4;
  const int mrow = lane & 15;   // M (row) for the A operand

#if HAVE_WMMA_F32X4
  v8f S0 = {}, S2 = {}, T0 = {}, T1 = {}, T2 = {};
#pragma unroll
  for (int cch = 0; cch < 5; ++cch) {
    v2f aS, aD, b0, b1, b2;
#pragma unroll
    for (int j = 0; j < 2; ++j) {
      int r = 4 * cch + 2 * half + j;      // K index in this chunk
      aS[j] = band_smooth(mrow, r);
      aD[j] = band_diff(mrow, r);
      b0[j] = g[r][nn + 0];                // rows >= 18 are zero padded
      b1[j] = g[r][nn + 1];
      b2[j] = g[r][nn + 2];
    }
    // (neg_a, A, neg_b, B, c_mod, C, reuse_a, reuse_b)
    S0 = __builtin_amdgcn_wmma_f32_16x16x4_f32(false, aS, false, b0, (short)0, S0, false, false);
    S2 = __builtin_amdgcn_wmma_f32_16x16x4_f32(false, aS, false, b2, (short)0, S2, false, false);
    T0 = __builtin_amdgcn_wmma_f32_16x16x4_f32(false, aD, false, b0, (short)0, T0, false, false);
    T1 = __builtin_amdgcn_wmma_f32_16x16x4_f32(false, aD, false, b1, (short)0, T1, false, false);
    T2 = __builtin_amdgcn_wmma_f32_16x16x4_f32(false, aD, false, b2, (short)0, T2, false, false);
  }
#pragma unroll
  for (int v = 0; v < 8; ++v) {
    float gxv = S2[v] - S0[v];
    float gyv = T0[v] + 2.0f * T1[v] + T2[v];
    float gr  = sqrtf(gxv * gxv + gyv * gyv);
    int m = v + 8 * half;                  // C/D layout: VGPR v, lane half
    size_t p = (size_t)b * HW + (size_t)(ty + m) * Ww + (tx + nn);
    grad_out[p] = gr;
    float g2 = gr * gr;
    wgrad[p] = g2 * g2 * GRADW;
  }
#else
  // scalar fallback: each lane does 8 rows of its column
#pragma unroll
  for (int v = 0; v < 8; ++v) {
    int m = v + 8 * half;
    int r = m + 1, c = nn + 1;
    float gxv = (g[r-1][c+1] + 2.0f*g[r][c+1] + g[r+1][c+1])
              - (g[r-1][c-1] + 2.0f*g[r][c-1] + g[r+1][c-1]);
    float gyv = (g[r+1][c-1] + 2.0f*g[r+1][c] + g[r+1][c+1])
              - (g[r-1][c-1] + 2.0f*g[r-1][c] + g[r-1][c+1]);
    float gr = sqrtf(gxv * gxv + gyv * gyv);
    size_t p = (size_t)b * HW + (size_t)(ty + m) * Ww + (tx + nn);
    grad_out[p] = gr;
    float g2 = gr * gr;
    wgrad[p] = g2 * g2 * GRADW;
  }
#endif
}

// ---------------------------------------------------------------------------
// Kernel 2: per (batch, cluster) 20x20 window argmin of grad (first-occurrence
// tie break, matching jnp.argmin). wave32 shfl reduce + cross-wave LDS reduce.
// ---------------------------------------------------------------------------
__global__ __launch_bounds__(128) void k_centroids(const float* __restrict__ grad,
                                                   int* __restrict__ cent_i,
                                                   float* __restrict__ cent_f) {
  const int bc  = blockIdx.x;            // b*64 + cid
  const int b   = bc >> 6, cid = bc & 63;
  const int ci  = cid >> 3, cj = cid & 7;
  const int y0  = ci * 64 + 32 - NEIGHR;
  const int x0  = cj * 64 + 32 - NEIGHR;
  const float* gb = grad + (size_t)b * HW;

  const int t = threadIdx.x;
  float bv = __builtin_inff();
  int   bi = WINS * WINS;
  for (int i = t; i < WINS * WINS; i += 128) {
    int wy = i / WINS, wx = i - wy * WINS;
    float v = gb[(size_t)(y0 + wy) * Ww + (x0 + wx)];
    if (v < bv) { bv = v; bi = i; }   // increasing i => first occurrence kept
  }
  for (int off = 16; off > 0; off >>= 1) {
    float ov = __shfl_down(bv, off, 32);
    int   oi = __shfl_down(bi, off, 32);
    if (ov < bv || (ov == bv && oi < bi)) { bv = ov; bi = oi; }
  }
  __shared__ float sv[4];
  __shared__ int   si[4];
  if ((t & 31) == 0) { sv[t >> 5] = bv; si[t >> 5] = bi; }
  __syncthreads();
  if (t == 0) {
    for (int w = 1; w < 4; ++w)
      if (sv[w] < bv || (sv[w] == bv && si[w] < bi)) { bv = sv[w]; bi = si[w]; }
    int cy = y0 + bi / WINS;
    int cx = x0 + bi - (bi / WINS) * WINS;
    cent_i[bc * 2 + 0] = cy;
    cent_i[bc * 2 + 1] = cx;
    cent_f[bc * 2 + 0] = (float)cy;
    cent_f[bc * 2 + 1] = (float)cx;
  }
}

// ---------------------------------------------------------------------------
// Kernel 3/4: init dist=inf, mask=-1; seed centroid pixels.
// ---------------------------------------------------------------------------
__global__ __launch_bounds__(256) void k_init(float* __restrict__ dist,
                                              int* __restrict__ mask) {
  size_t i = (size_t)blockIdx.x * 256 + threadIdx.x;
  dist[i] = __builtin_inff();
  mask[i] = -1;
}

__global__ __launch_bounds__(256) void k_seed(const int* __restrict__ cent_i,
                                              float* __restrict__ dist,
                                              int* __restrict__ mask) {
  int i = blockIdx.x * 256 + threadIdx.x;
  if (i < BATCH * NCLUST) {
    int b = i >> 6, cid = i & 63;
    int cy = cent_i[i * 2 + 0], cx = cent_i[i * 2 + 1];
    size_t p = (size_t)b * HW + (size_t)cy * Ww + cx;
    dist[p] = 0.0f;
    mask[p] = cid;
  }
}

// ---------------------------------------------------------------------------
// Kernel 5: one directional min-plus pass. Neighbor dist/mask gathered
// asynchronously into LDS (per-lane async op, handles wraparound gather).
// ---------------------------------------------------------------------------
__global__ __launch_bounds__(256) void k_prop(const float* __restrict__ x,
                                              const float* __restrict__ wgrad,
                                              const float* __restrict__ din,
                                              const int* __restrict__ min_,
                                              float* __restrict__ dout,
                                              int* __restrict__ mout,
                                              int dy, int dx) {
  __shared__ float sdq[256];
  __shared__ int   smq[256];
  const int tid = threadIdx.x;
  size_t gi = (size_t)blockIdx.x * 256 + tid;   // grid is exact: BHW/256

  int b  = (int)(gi >> 18);          // HW = 2^18
  int p  = (int)(gi & (HW - 1));
  int y  = p >> 9, xc = p & (Ww - 1);
  int qy = (y - dy) & (Hh - 1);
  int qx = (xc - dx) & (Ww - 1);
  int q  = (qy << 9) | qx;
  size_t base = (size_t)b * HW;

  async_gather_b32(&sdq[tid], din + base + q);
  async_gather_b32(&smq[tid], min_ + base + q);

  // overlap: compute penalty while async copies are in flight
  const float* xb = x + (size_t)b * 3 * HW;
  float pen = wgrad[base + p];
  float c0 = fabsf(xb[p]          - xb[q]);
  float c1 = fabsf(xb[HW + p]     - xb[HW + q]);
  float c2 = fabsf(xb[2 * HW + p] - xb[2 * HW + q]);
  pen += (c0 + c1 + c2) * COLORW;

  float dcur = din[base + p];
  int   mcur = min_[base + p];

  wait_async0();
  float dq = sdq[tid];
  int   mq = smq[tid];

  float nd = dq + pen;
  bool upd = nd < dcur;
  dout[base + p] = upd ? nd : dcur;
  mout[base + p] = upd ? mq : mcur;
}

// ---------------------------------------------------------------------------
// Kernel 6: int mask -> float output section.
// ---------------------------------------------------------------------------
__global__ __launch_bounds__(256) void k_finalize(const int* __restrict__ mask,
                                                  float* __restrict__ out) {
  size_t i = (size_t)blockIdx.x * 256 + threadIdx.x;
  out[i] = (float)mask[i];
}

// ---------------------------------------------------------------------------
extern "C" void kernel_launch(void* const* d_in, const int* in_sizes, int n_in,
                              void* d_out, int out_size, void* d_ws, size_t ws_size,
                              hipStream_t stream) {
  const float* x = (const float*)d_in[0];

  float* out      = (float*)d_out;
  float* grad_out = out;                                  // BHW floats
  float* cent_f   = out + (size_t)BHW;                    // B*64*2 floats
  float* mask_out = cent_f + (size_t)BATCH * NCLUST * 2;  // BHW floats

  char* ws = (char*)d_ws;
  float* wgrad  = (float*)ws;  ws += (size_t)BHW * 4;
  float* dist0  = (float*)ws;  ws += (size_t)BHW * 4;
  float* dist1  = (float*)ws;  ws += (size_t)BHW * 4;
  int*   mask0  = (int*)ws;    ws += (size_t)BHW * 4;
  int*   mask1  = (int*)ws;    ws += (size_t)BHW * 4;
  int*   cent_i = (int*)ws;

  dim3 gpre(Ww / 16, Hh / 16, BATCH);
  k_preprocess<<<gpre, 32, 0, stream>>>(x, grad_out, wgrad);

  k_centroids<<<BATCH * NCLUST, 128, 0, stream>>>(grad_out, cent_i, cent_f);

  k_init<<<BHW / 256, 256, 0, stream>>>(dist0, mask0);
  k_seed<<<(BATCH * NCLUST + 255) / 256, 256, 0, stream>>>(cent_i, dist0, mask0);

  const int DY[4] = {-1, 1, 0, 0};
  const int DX[4] = {0, 0, -1, 1};
  float *da = dist0, *db = dist1;
  int   *ma = mask0, *mb = mask1;
  for (int it = 0; it < NITER; ++it) {
    for (int d = 0; d < 4; ++d) {
      k_prop<<<BHW / 256, 256, 0, stream>>>(x, wgrad, da, ma, db, mb, DY[d], DX[d]);
      float* tf = da; da = db; db = tf;
      int*   ti = ma; ma = mb; mb = ti;
    }
  }
  k_finalize<<<BHW / 256, 256, 0, stream>>>(ma, mask_out);
}